// CausalSelfAttention_89146341196357
// MI455X (gfx1250) — compile-verified
//
#include <hip/hip_runtime.h>
#include <math.h>

typedef float v2f __attribute__((ext_vector_type(2)));
typedef float v8f __attribute__((ext_vector_type(8)));

// D = A(16x4) * B(4x16) + C, full fp32 WMMA (CDNA5)
static __device__ __forceinline__ v8f wmma4(v2f a, v2f b, v8f c) {
  return __builtin_amdgcn_wmma_f32_16x16x4_f32(false, a, false, b, (short)0, c,
                                               false, false);
}

// CDNA5 async copy global->LDS (16B per lane), tracked by ASYNCcnt.
static __device__ __forceinline__ void async_load_b128(void* lds_ptr,
                                                       const float* gptr) {
  uint32_t l = (uint32_t)(uintptr_t)lds_ptr;  // low 32 bits = LDS byte offset
  asm volatile("global_load_async_to_lds_b128 %0, %1, off" ::"v"(l), "v"(gptr)
               : "memory");
}
// ASYNC loads complete in order: waiting ASYNCcnt<=N leaves only the newest N
// (the prefetched panel) in flight while the current panel is guaranteed done.
static __device__ __forceinline__ void wait_async0() {
  asm volatile("s_wait_asynccnt 0x0" ::: "memory");
}
static __device__ __forceinline__ void wait_async8() {
  asm volatile("s_wait_asynccnt 0x8" ::: "memory");
}
static __device__ __forceinline__ void wait_async4() {
  asm volatile("s_wait_asynccnt 0x4" ::: "memory");
}

static __device__ __forceinline__ float rmax16(float x) {
#pragma unroll
  for (int m = 1; m < 16; m <<= 1) x = fmaxf(x, __shfl_xor(x, m, 16));
  return x;
}
static __device__ __forceinline__ float rsum16(float x) {
#pragma unroll
  for (int m = 1; m < 16; m <<= 1) x += __shfl_xor(x, m, 16);
  return x;
}

// ---------------------------------------------------------------------------
// C[M,N] = A[M,K] (row-major) * B[N,K]^T (row-major).
// Block tile 128x128 (8 waves, each 32x64). K staged in double-buffered LDS
// panels of 32 filled by async-to-LDS copies (next panel overlaps current
// panel's 64 WMMAs). Stride-36 padding: conflict-free ds_load_b64 fragment
// reads, 16B-aligned async stores.
// ---------------------------------------------------------------------------
__global__ __launch_bounds__(256) void gemm_rt_kernel(
    const float* __restrict__ A, const float* __restrict__ B,
    float* __restrict__ C, int M, int N, int K, int tilesN) {
  __shared__ float lsA[2][128 * 36];
  __shared__ float lsB[2][128 * 36];

  const int tid = threadIdx.x;
  const int lane = tid & 31;
  const int wv = tid >> 5;
  const int l16 = lane & 15;
  const int hf = lane >> 4;

  const int tm = blockIdx.x / tilesN;
  const int tn = blockIdx.x - tm * tilesN;
  const int m0 = tm * 128, n0 = tn * 128;
  const int wm = (wv >> 1) * 32;  // wave row offset inside block tile
  const int wn = (wv & 1) * 64;   // wave col offset inside block tile

  v8f acc[2][4];
#pragma unroll
  for (int i = 0; i < 2; ++i)
#pragma unroll
    for (int j = 0; j < 4; ++j)
#pragma unroll
      for (int r = 0; r < 8; ++r) acc[i][j][r] = 0.0f;

  // Panel fill: 8 async b128 per thread (4 for A, 4 for B).
  auto fill = [&](int kp, int buf) {
#pragma unroll
    for (int it = 0; it < 4; ++it) {
      const int f = tid + it * 256;
      const int row = f >> 3;
      const int kc = (f & 7) * 4;
      async_load_b128(&lsA[buf][row * 36 + kc],
                      A + (size_t)(m0 + row) * K + kp + kc);
      async_load_b128(&lsB[buf][row * 36 + kc],
                      B + (size_t)(n0 + row) * K + kp + kc);
    }
  };

  fill(0, 0);  // prologue
  const int npanels = K / 32;
  for (int pi = 0; pi < npanels; ++pi) {
    const int buf = pi & 1;
    if (pi + 1 < npanels) {
      fill((pi + 1) * 32, buf ^ 1);  // prefetch next panel into spare buffer
      wait_async8();                 // drain current panel only
    } else {
      wait_async0();
    }
    __syncthreads();  // current panel visible to all waves

    const float* aw = &lsA[buf][(wm + l16) * 36 + 2 * hf];
    const float* bw = &lsB[buf][(wn + l16) * 36 + 2 * hf];
#pragma unroll
    for (int k0 = 0; k0 < 32; k0 += 4) {
      v2f av[2], bv[4];
#pragma unroll
      for (int i = 0; i < 2; ++i) av[i] = *(const v2f*)(aw + i * 16 * 36 + k0);
#pragma unroll
      for (int j = 0; j < 4; ++j) bv[j] = *(const v2f*)(bw + j * 16 * 36 + k0);
#pragma unroll
      for (int i = 0; i < 2; ++i)
#pragma unroll
        for (int j = 0; j < 4; ++j) acc[i][j] = wmma4(av[i], bv[j], acc[i][j]);
    }
    __syncthreads();  // all waves done with this buffer before it is refilled
  }

#pragma unroll
  for (int i = 0; i < 2; ++i)
#pragma unroll
    for (int j = 0; j < 4; ++j)
#pragma unroll
      for (int r = 0; r < 8; ++r) {
        const int row = m0 + wm + i * 16 + r + 8 * hf;
        const int col = n0 + wn + j * 16 + l16;
        C[(size_t)row * N + col] = acc[i][j][r];
      }
}

// ---------------------------------------------------------------------------
// Llama-style half-rotation RoPE, in place. x: [rows, nh*128]; T = 2048.
// ---------------------------------------------------------------------------
__global__ void rope_kernel(float* __restrict__ x, const int* __restrict__ sp,
                            int nh, int total) {
  const int idx = blockIdx.x * blockDim.x + threadIdx.x;
  if (idx >= total) return;
  const int d = idx & 63;
  const int t1 = idx >> 6;
  const int hh = t1 % nh;
  const int row = t1 / nh;
  const int t = (row & 2047) + sp[0];
  const float inv = powf(10000.0f, -(float)d * (1.0f / 64.0f));
  const float ang = (float)t * inv;
  const float c = cosf(ang), s = sinf(ang);
  float* p = x + (size_t)row * (nh * 128) + hh * 128 + d;
  const float x1 = p[0], x2 = p[64];
  p[0] = x1 * c - x2 * s;
  p[64] = x2 * c + x1 * s;
}

// ---------------------------------------------------------------------------
// Non-masked flash attention (streaming softmax), GQA (16 Q heads / 4 KV).
// Workgroup: 8 waves x 16 query rows = 128 rows of one (b,h).
// K/V 16x128 tiles double-buffered in LDS via async-to-LDS (next tile's
// copies overlap current tile's WMMA + softmax); Q fragments in registers.
// ---------------------------------------------------------------------------
__global__ __launch_bounds__(256) void attn_kernel(const float* __restrict__ q,
                                                   const float* __restrict__ k,
                                                   const float* __restrict__ v,
                                                   float* __restrict__ o) {
  __shared__ float lk[2][16 * 128];
  __shared__ float lv[2][16 * 128];
  __shared__ float lp[8 * 256];  // per-wave 16x16 P tile

  const int T = 2048;
  const int lane = threadIdx.x & 31;
  const int wv = threadIdx.x >> 5;
  const int l16 = lane & 15;
  const int hf = lane >> 4;

  const int rb = blockIdx.x & 15;  // T/128 = 16 row blocks
  const int bh = blockIdx.x >> 4;  // b*16 + h
  const int b = bh >> 4;
  const int h = bh & 15;
  const int g = h >> 2;  // kv head

  const int q0 = rb * 128 + wv * 16;

  // Preload Q (16 rows x 128 dims) as 32 A-fragments.
  v2f qa[32];
  {
    const float* qp = q + (size_t)(b * T + q0 + l16) * 2048 + h * 128 + 2 * hf;
#pragma unroll
    for (int kk = 0; kk < 32; ++kk) qa[kk] = *(const v2f*)(qp + kk * 4);
  }

  v8f oacc[8];
#pragma unroll
  for (int n = 0; n < 8; ++n)
#pragma unroll
    for (int r = 0; r < 8; ++r) oacc[n][r] = 0.0f;
  float mrow[8], lrow[8];
#pragma unroll
  for (int j = 0; j < 8; ++j) {
    mrow[j] = -1e30f;
    lrow[j] = 0.0f;
  }

  const float scale = 0.08838834764831845f;  // 1/sqrt(128)
  const size_t kvbase = (size_t)(b * T) * 512 + g * 128;
  float* myp = lp + wv * 256;

  // K/V tile fill: 4 async b128 per thread (2 for K, 2 for V).
  auto fillkv = [&](int key0, int buf) {
#pragma unroll
    for (int it = 0; it < 2; ++it) {
      const int j4 = threadIdx.x + it * 256;  // float4 index 0..511
      const int row = j4 >> 5;
      const int col = (j4 & 31) * 4;
      const size_t base = kvbase + (size_t)(key0 + row) * 512 + col;
      async_load_b128(&lk[buf][row * 128 + col], k + base);
      async_load_b128(&lv[buf][row * 128 + col], v + base);
    }
  };

  fillkv(0, 0);  // prologue
  for (int kt = 0; kt < T / 16; ++kt) {
    const int buf = kt & 1;
    if (kt + 1 < T / 16) {
      fillkv((kt + 1) * 16, buf ^ 1);  // prefetch next tile into spare buffer
      wait_async4();                   // drain current tile only
    } else {
      wait_async0();
    }
    __syncthreads();  // current tile visible to all waves

    // S = Q * K^T for this wave's 16x16 tile (K from LDS).
    v8f s;
#pragma unroll
    for (int r = 0; r < 8; ++r) s[r] = 0.0f;
#pragma unroll
    for (int kk = 0; kk < 32; ++kk) {
      v2f bv = *(const v2f*)&lk[buf][l16 * 128 + kk * 4 + 2 * hf];
      s = wmma4(qa[kk], bv, s);
    }

    // Streaming softmax. Row r=j+8*hf lives in 16 lanes of this half-wave.
    float p8[8], alpha[8];
#pragma unroll
    for (int j = 0; j < 8; ++j) {
      const float sv = s[j] * scale;
      const float mt = rmax16(sv);
      const float mn = fmaxf(mrow[j], mt);
      const float pj = expf(sv - mn);
      const float rs = rsum16(pj);
      alpha[j] = expf(mrow[j] - mn);
      lrow[j] = lrow[j] * alpha[j] + rs;
      mrow[j] = mn;
      p8[j] = pj;
    }
#pragma unroll
    for (int n = 0; n < 8; ++n)
#pragma unroll
      for (int j = 0; j < 8; ++j) oacc[n][j] *= alpha[j];

    // Re-shape P (D-layout) -> A-layout via per-wave LDS bounce.
#pragma unroll
    for (int j = 0; j < 8; ++j) myp[(j + 8 * hf) * 16 + l16] = p8[j];
    __syncthreads();

    // O += P * V  (K-dim = 16 keys, split into 4 WMMA k-slices).
#pragma unroll
    for (int kk = 0; kk < 4; ++kk) {
      v2f pa = *(const v2f*)&myp[l16 * 16 + kk * 4 + 2 * hf];
#pragma unroll
      for (int n = 0; n < 8; ++n) {
        v2f bv;
        bv[0] = lv[buf][(kk * 4 + 2 * hf) * 128 + n * 16 + l16];
        bv[1] = lv[buf][(kk * 4 + 2 * hf + 1) * 128 + n * 16 + l16];
        oacc[n] = wmma4(pa, bv, oacc[n]);
      }
    }
    __syncthreads();  // all waves done with this buffer before it is refilled
  }

  float invl[8];
#pragma unroll
  for (int j = 0; j < 8; ++j) invl[j] = 1.0f / lrow[j];
#pragma unroll
  for (int n = 0; n < 8; ++n)
#pragma unroll
    for (int j = 0; j < 8; ++j) {
      const int row = q0 + j + 8 * hf;
      o[(size_t)(b * T + row) * 2048 + h * 128 + n * 16 + l16] =
          oacc[n][j] * invl[j];
    }
}

// ---------------------------------------------------------------------------
extern "C" void kernel_launch(void* const* d_in, const int* in_sizes, int n_in,
                              void* d_out, int out_size, void* d_ws,
                              size_t ws_size, hipStream_t stream) {
  (void)in_sizes;
  (void)n_in;
  (void)out_size;
  (void)ws_size;
  const float* x = (const float*)d_in[0];
  const float* Wq = (const float*)d_in[1];
  const float* Wk = (const float*)d_in[2];
  const float* Wv = (const float*)d_in[3];
  const float* Wo = (const float*)d_in[4];
  const int* sp = (const int*)d_in[5];

  // Workspace layout (floats): q[4096*2048] k[4096*512] v[4096*512] att[4096*2048]
  float* q = (float*)d_ws;
  float* kb = q + (size_t)4096 * 2048;
  float* vb = kb + (size_t)4096 * 512;
  float* at = vb + (size_t)4096 * 512;
  float* out = (float*)d_out;

  // QKV projections: one block per 128x128 C tile.
  gemm_rt_kernel<<<32 * 16, 256, 0, stream>>>(x, Wq, q, 4096, 2048, 2048, 16);
  gemm_rt_kernel<<<32 * 4, 256, 0, stream>>>(x, Wk, kb, 4096, 512, 2048, 4);
  gemm_rt_kernel<<<32 * 4, 256, 0, stream>>>(x, Wv, vb, 4096, 512, 2048, 4);
  // RoPE on q and k
  rope_kernel<<<(4096 * 16 * 64) / 256, 256, 0, stream>>>(q, sp, 16,
                                                          4096 * 16 * 64);
  rope_kernel<<<(4096 * 4 * 64) / 256, 256, 0, stream>>>(kb, sp, 4,
                                                         4096 * 4 * 64);
  // Attention (512 workgroups: 2 batches x 16 heads x 16 row-blocks)
  attn_kernel<<<512, 256, 0, stream>>>(q, kb, vb, at);
  // Output projection
  gemm_rt_kernel<<<32 * 16, 256, 0, stream>>>(at, Wo, out, 4096, 2048, 2048,
                                              16);
}